// KoopmanAutoencoder_55052890800320
// MI455X (gfx1250) — compile-verified
//
#include <hip/hip_runtime.h>
#include <hip/hip_bf16.h>

typedef __attribute__((ext_vector_type(2))) float v2f;
typedef __attribute__((ext_vector_type(8))) float v8f;

#define B_    16
#define N_    128
#define T_    1024
#define SW_   64
#define F_    16
#define LAT_  32
#define ELAT_ 64
#define NC_   8
#define M_    20
#define E_    8128
#define DT_   0.1f
#define EPS_  1e-5f
#define KSTEPS 4

// ---------------------------------------------------------------------------
// Tensor Data Mover staging (CDNA5).  2D tile -> LDS with hardware row
// padding (pad_amount = 1 DWORD per row) so the staged tile lands with the
// (D+1) stride needed for bank-conflict-free WMMA fragment reads.
// ---------------------------------------------------------------------------
#if defined(__HIP_DEVICE_COMPILE__)
#if __has_builtin(__builtin_amdgcn_tensor_load_to_lds)
#define HAVE_TDM 1
#endif
#endif
#ifndef HAVE_TDM
#define HAVE_TDM 0
#endif

#if HAVE_TDM
typedef __attribute__((ext_vector_type(4))) unsigned int u32x4;
typedef __attribute__((ext_vector_type(8))) int          i32x8;
typedef __attribute__((ext_vector_type(4))) int          i32x4;

// tile_w in DWORDs (row length), tile_h rows, stride_w DWORDs between rows
// in global memory, pad_code: pad 1 DWORD after every 2^(pad_code+1) DWORDs.
__device__ inline void tdm_load_2d(unsigned lds_off, const float* gsrc,
                                   int tile_w, int tile_h, long stride_w,
                                   int pad_code)
{
  const unsigned long long ga = (unsigned long long)gsrc;
  u32x4 g0;
  g0[0] = 1u;                                              // count=1, user D#
  g0[1] = lds_off;                                         // lds_addr
  g0[2] = (unsigned)(ga & 0xFFFFFFFFull);                  // global_addr lo
  g0[3] = (unsigned)((ga >> 32) & 0x1FFFFFFull) | (2u << 30); // addr hi | type=2
  i32x8 g1;
  g1[0] = (2 << 16) | (1 << 20) | (pad_code << 22);        // 4B elems, pad_en
  g1[1] = (tile_w & 0xFFFF) << 16;                         // tensor_dim0 lo16
  g1[2] = ((tile_h & 0xFFFF) << 16);                       // tdim0 hi=0 | tdim1 lo
  g1[3] = ((tile_w & 0xFFFF) << 16);                       // tdim1 hi=0 | tile_dim0
  g1[4] = (tile_h & 0xFFFF);                               // tile_dim1 (dim2=0)
  g1[5] = (int)(stride_w & 0xFFFFFFFF);                    // dim0_stride lo
  g1[6] = (int)((stride_w >> 32) & 0xFFFF);                // dim0_stride hi
  g1[7] = 0;
  const i32x4 z4 = {};
  const i32x8 z8 = {};
  __builtin_amdgcn_tensor_load_to_lds(g0, g1, z4, z4, z8, 0);
}
#endif

// ---------------------------------------------------------------------------
// fp32 WMMA 16x16x4 fragment helpers.  A 16x4 = float2/lane (lanes 0-15 hold
// K={0,1}, lanes 16-31 K={2,3}); B 4x16 mirrored.  Dual accumulators split
// the serial WMMA dependency chain (K always a multiple of 8 here).
// ---------------------------------------------------------------------------
__device__ inline v8f wmma_tile_ab(const float* __restrict__ A, int lda,
                                   const float* __restrict__ B, int ldb,
                                   int K, int lane) {
  const int half = lane >> 4, l16 = lane & 15;
  const float* ap = A + l16 * lda + 2 * half;
  const float* bp = B + (2 * half) * ldb + l16;
  const int bstep = 4 * ldb;
  v8f acc0 = {}, acc1 = {};
  for (int k = 0; k < K; k += 8) {
    v2f a0; a0[0] = ap[0]; a0[1] = ap[1];
    v2f b0; b0[0] = bp[0]; b0[1] = bp[ldb];
    acc0 = __builtin_amdgcn_wmma_f32_16x16x4_f32(false, a0, false, b0,
                                                 (short)0, acc0, false, false);
    v2f a1; a1[0] = ap[4]; a1[1] = ap[5];
    v2f b1; b1[0] = bp[bstep]; b1[1] = bp[bstep + ldb];
    acc1 = __builtin_amdgcn_wmma_f32_16x16x4_f32(false, a1, false, b1,
                                                 (short)0, acc1, false, false);
    ap += 8; bp += 2 * bstep;
  }
  return acc0 + acc1;
}

// Ac @ Ac^T tile: B-fragment at column-tile tj equals the A-fragment pattern
// applied at rows tj*16 (symmetric operand trick).
__device__ inline v8f wmma_tile_aat(const float* __restrict__ Ai,
                                    const float* __restrict__ Aj,
                                    int lda, int K, int lane) {
  const int half = lane >> 4, l16 = lane & 15;
  const float* ap = Ai + l16 * lda + 2 * half;
  const float* bp = Aj + l16 * lda + 2 * half;
  v8f acc0 = {}, acc1 = {};
  for (int k = 0; k < K; k += 8) {
    v2f a0; a0[0] = ap[0]; a0[1] = ap[1];
    v2f b0; b0[0] = bp[0]; b0[1] = bp[1];
    acc0 = __builtin_amdgcn_wmma_f32_16x16x4_f32(false, a0, false, b0,
                                                 (short)0, acc0, false, false);
    v2f a1; a1[0] = ap[4]; a1[1] = ap[5];
    v2f b1; b1[0] = bp[4]; b1[1] = bp[5];
    acc1 = __builtin_amdgcn_wmma_f32_16x16x4_f32(false, a1, false, b1,
                                                 (short)0, acc1, false, false);
    ap += 8; bp += 8;
  }
  return acc0 + acc1;
}

// ---------------------------------------------------------------------------
// Wide GEMM: one wave computes a 16 x (16*NT) strip with NT independent
// accumulators; each A fragment is reused across NT WMMAs.  Full tiles only.
// ---------------------------------------------------------------------------
template <int NT>
__global__ void __launch_bounds__(32) gemm_wmma_wide(
    const float* __restrict__ A, int lda, long aBatch,
    const float* __restrict__ Bm, int ldb, long bBatch,
    const float* __restrict__ bias, long biasBatch,
    float* __restrict__ C, int ldc, long cBatch,
    int K, int relu)
{
  const int lane = threadIdx.x;
  const int half = lane >> 4, l16 = lane & 15;
  const int row0 = blockIdx.y * 16, col0 = blockIdx.x * (16 * NT);
  const int bat  = blockIdx.z;

  const float* ap = A + (long)bat * aBatch + (long)(row0 + l16) * lda + 2 * half;
  const float* bp = Bm + (long)bat * bBatch + (long)(2 * half) * ldb + col0 + l16;
  const float* bi = bias + (long)bat * biasBatch + col0 + l16;
  float*       c  = C + (long)bat * cBatch + col0 + l16;

  v8f acc[NT];
  const v8f zero = {};
  for (int t = 0; t < NT; ++t) acc[t] = zero;

  for (int k = 0; k < K; k += 4) {
    v2f af; af[0] = ap[0]; af[1] = ap[1];
    for (int t = 0; t < NT; ++t) {
      v2f bf; bf[0] = bp[t * 16]; bf[1] = bp[ldb + t * 16];
      acc[t] = __builtin_amdgcn_wmma_f32_16x16x4_f32(false, af, false, bf,
                                                     (short)0, acc[t],
                                                     false, false);
    }
    ap += 4; bp += 4L * ldb;
  }
  for (int t = 0; t < NT; ++t) {
    const float bv = bi[t * 16];
    for (int v = 0; v < 8; ++v) {
      const int row = row0 + v + 8 * half;
      float val = acc[t][v] + bv;
      if (relu) val = fmaxf(val, 0.f);
      c[(long)row * ldc + t * 16] = val;
    }
  }
}

// ---------------------------------------------------------------------------
// Generic GEMM (ragged Ncols via address clamping — branch-free; clamped
// columns never stored and never contaminate valid columns).
// ---------------------------------------------------------------------------
__global__ void __launch_bounds__(32) gemm_wmma(
    const float* __restrict__ A, int lda, long aBatch,
    const float* __restrict__ Bm, int ldb, long bBatch,
    const float* __restrict__ bias, long biasBatch,
    float* __restrict__ C, int ldc, long cBatch,
    int Ncols, int K, int relu)
{
  const int lane = threadIdx.x;
  const int half = lane >> 4, l16 = lane & 15;
  const int row0 = blockIdx.y * 16, col0 = blockIdx.x * 16;
  const int bat  = blockIdx.z;

  const int  col   = col0 + l16;
  const bool colok = (col < Ncols);
  const int  colc  = colok ? col : (Ncols - 1);

  const float* ap = A + (long)bat * aBatch + (long)(row0 + l16) * lda + 2 * half;
  const float* bp = Bm + (long)bat * bBatch + (long)(2 * half) * ldb + colc;
  const float* bi = bias + (long)bat * biasBatch;
  float*       c  = C + (long)bat * cBatch;
  const int bstep = 4 * ldb;

  v8f acc0 = {}, acc1 = {};
  for (int k = 0; k < K; k += 8) {
    v2f a0; a0[0] = ap[0]; a0[1] = ap[1];
    v2f b0; b0[0] = bp[0]; b0[1] = bp[ldb];
    acc0 = __builtin_amdgcn_wmma_f32_16x16x4_f32(false, a0, false, b0,
                                                 (short)0, acc0, false, false);
    v2f a1; a1[0] = ap[4]; a1[1] = ap[5];
    v2f b1; b1[0] = bp[bstep]; b1[1] = bp[bstep + ldb];
    acc1 = __builtin_amdgcn_wmma_f32_16x16x4_f32(false, a1, false, b1,
                                                 (short)0, acc1, false, false);
    ap += 8; bp += 2 * bstep;
  }
  const v8f acc = acc0 + acc1;
  const float bv = bi[colc];
  if (colok) {
    for (int v = 0; v < 8; ++v) {
      const int row = row0 + v + 8 * half;
      float val = acc[v] + bv;
      if (relu) val = fmaxf(val, 0.f);
      c[(long)row * ldc + col] = val;
    }
  }
}

// ---------------------------------------------------------------------------
// Fused pearson + per-edge MLP.  One block per (batch) 128 x D slice; only
// the 36 upper-triangular 16x16 tiles of Ac@Ac^T are computed.  Input tile
// staged by the TDM with hardware row padding (stride D+1).
// ---------------------------------------------------------------------------
template <int D, bool PER_M>
__global__ void __launch_bounds__(256) pearson_edge_kernel(
    const float* __restrict__ src,
    const float* __restrict__ W1, const float* __restrict__ b1,
    const float* __restrict__ W2, const float* __restrict__ b2,
    float* __restrict__ outp)
{
  __shared__ float Ac[N_ * (D + 1)];
  __shared__ float sdev[N_];
  __shared__ float w1s[16], b1s[16], w2s[16];
  __shared__ float b2s;

  const int bid  = blockIdx.x;
  const int tid  = threadIdx.x;
  const int wave = tid >> 5, lane = tid & 31;
  const int half = lane >> 4, l16 = lane & 15;

  const int widx = PER_M ? (bid / B_) : 0;
  if (tid < 16) {
    w1s[tid] = W1[widx * 16 + tid];
    b1s[tid] = b1[widx * 16 + tid];
    w2s[tid] = W2[widx * 16 + tid];
  }
  if (tid == 0) b2s = b2[widx];

  const float* A = src + (long)bid * N_ * D;
#if HAVE_TDM
  if (tid < 32) {   // one wave issues the DMA (TDM ignores EXEC)
    tdm_load_2d((unsigned)(unsigned long long)(void*)Ac, A,
                D, N_, D, (D == 64) ? 5 : 4);
    __builtin_amdgcn_s_wait_tensorcnt(0);
  }
#else
  for (int i = tid; i < N_ * D; i += 256) {
    const int n = i / D, d = i % D;
    Ac[n * (D + 1) + d] = A[n * D + d];
  }
#endif
  __syncthreads();

  if (tid < N_) {
    float m = 0.f;
    for (int d = 0; d < D; ++d) m += Ac[tid * (D + 1) + d];
    m /= (float)D;
    float var = 0.f;
    for (int d = 0; d < D; ++d) {
      const float v = Ac[tid * (D + 1) + d] - m;
      Ac[tid * (D + 1) + d] = v;
      var += v * v;
    }
    sdev[tid] = sqrtf(var / (float)(D - 1));
  }
  __syncthreads();

  for (int t = wave; t < 36; t += 8) {
    int ti = 0, rem = t;
    while (rem >= 8 - ti) { rem -= 8 - ti; ++ti; }
    const int tj = ti + rem;

    v8f acc = wmma_tile_aat(&Ac[ti * 16 * (D + 1)], &Ac[tj * 16 * (D + 1)],
                            D + 1, D, lane);

    for (int v = 0; v < 8; ++v) {
      const int i = ti * 16 + v + 8 * half;
      const int j = tj * 16 + l16;
      if (j > i) {
        const float cov = acc[v] / (float)(D - 1);
        const float ge  = cov / (sdev[i] * sdev[j] + EPS_);
        float o = b2s;
        for (int h = 0; h < 16; ++h)
          o += fmaxf(ge * w1s[h] + b1s[h], 0.f) * w2s[h];
        const int e = i * (N_ - 1) - (i * (i - 1)) / 2 + (j - i - 1);
        if (PER_M) {
          const int m = bid / B_, b = bid % B_;
          outp[((long)b * E_ + e) * M_ + m] = o;
        } else {
          outp[(long)bid * E_ + e] = o;
        }
      }
    }
  }
}

// ---------------------------------------------------------------------------
// Fused encoder + decoder per (b,f): xw -> hidden -> y -> h2 -> x_recon,
// all tiles resident in LDS.  Input tile staged by the TDM (strided rows,
// hardware-padded to stride 65).
// ---------------------------------------------------------------------------
__global__ void __launch_bounds__(256) encdec_kernel(
    const float* __restrict__ x,
    const float* __restrict__ eeW1, const float* __restrict__ eeb1,
    const float* __restrict__ eeW2, const float* __restrict__ eeb2,
    const float* __restrict__ dW1,  const float* __restrict__ db1,
    const float* __restrict__ dW2,  const float* __restrict__ db2,
    float* __restrict__ y_out, float* __restrict__ xrec_out)
{
  __shared__ float xw[N_ * (SW_ + 1)];
  __shared__ float hid[N_ * (N_ + 1)];
  __shared__ float yt[N_ * (LAT_ + 1)];

  const int bf = blockIdx.x;
  const int b = bf / F_, f = bf % F_;
  const int tid = threadIdx.x;
  const int wave = tid >> 5, lane = tid & 31;
  const int half = lane >> 4, l16 = lane & 15;

#if HAVE_TDM
  if (tid < 32) {
    tdm_load_2d((unsigned)(unsigned long long)(void*)xw,
                x + ((long)b * N_) * T_ + f * SW_,
                SW_, N_, T_, 5);
    __builtin_amdgcn_s_wait_tensorcnt(0);
  }
#else
  for (int i = tid; i < N_ * SW_; i += 256) {
    const int n = i >> 6, s = i & 63;
    xw[n * (SW_ + 1) + s] = x[((long)b * N_ + n) * T_ + f * SW_ + s];
  }
#endif
  __syncthreads();

  // hidden = relu(xw @ eeW1 + eeb1)   (128x128, 64 tiles)
  for (int t = wave; t < 64; t += 8) {
    const int ti = t >> 3, tj = t & 7;
    v8f acc = wmma_tile_ab(&xw[ti * 16 * (SW_ + 1)], SW_ + 1,
                           &eeW1[tj * 16], N_, SW_, lane);
    const int col = tj * 16 + l16;
    const float bv = eeb1[col];
    for (int v = 0; v < 8; ++v)
      hid[(ti * 16 + v + 8 * half) * (N_ + 1) + col] = fmaxf(acc[v] + bv, 0.f);
  }
  __syncthreads();

  // y = hid @ eeW2 + eeb2   (128x32, 16 tiles) -> LDS + global
  for (int t = wave; t < 16; t += 8) {
    const int ti = t >> 1, tj = t & 1;
    v8f acc = wmma_tile_ab(&hid[ti * 16 * (N_ + 1)], N_ + 1,
                           &eeW2[tj * 16], LAT_, N_, lane);
    const int col = tj * 16 + l16;
    const float bv = eeb2[col];
    for (int v = 0; v < 8; ++v) {
      const int row = ti * 16 + v + 8 * half;
      const float val = acc[v] + bv;
      yt[row * (LAT_ + 1) + col] = val;
      y_out[(((long)b * F_ + f) * N_ + row) * LAT_ + col] = val;
    }
  }
  __syncthreads();

  // h2 = relu(y @ dW1 + db1)   (128x128)
  for (int t = wave; t < 64; t += 8) {
    const int ti = t >> 3, tj = t & 7;
    v8f acc = wmma_tile_ab(&yt[ti * 16 * (LAT_ + 1)], LAT_ + 1,
                           &dW1[tj * 16], N_, LAT_, lane);
    const int col = tj * 16 + l16;
    const float bv = db1[col];
    for (int v = 0; v < 8; ++v)
      hid[(ti * 16 + v + 8 * half) * (N_ + 1) + col] = fmaxf(acc[v] + bv, 0.f);
  }
  __syncthreads();

  // x_recon[b][n][f*64+s] = (h2 @ dW2 + db2)[n][s]   (128x64, 32 tiles)
  for (int t = wave; t < 32; t += 8) {
    const int ti = t >> 2, tj = t & 3;
    v8f acc = wmma_tile_ab(&hid[ti * 16 * (N_ + 1)], N_ + 1,
                           &dW2[tj * 16], SW_, N_, lane);
    const int col = tj * 16 + l16;
    const float bv = db2[col];
    for (int v = 0; v < 8; ++v) {
      const int row = ti * 16 + v + 8 * half;
      xrec_out[((long)b * N_ + row) * T_ + f * SW_ + col] = acc[v] + bv;
    }
  }
}

// ---------------------------------------------------------------------------
// vm: per (b,f) compute yU = U^T g, Koopman-advance for k=1..4, then
// g_next[k] = U @ y_next[k].
// ---------------------------------------------------------------------------
__global__ void __launch_bounds__(256) vm_kernel(
    const float* __restrict__ U,      // (B,E,M)
    const float* __restrict__ g,      // (B,F,E)
    const float* __restrict__ omega,  // (B,F,M)
    float* __restrict__ g_next)       // (K,B,F,E)
{
  __shared__ float red[256 * M_];
  __shared__ float yU[M_];
  __shared__ float yn[KSTEPS][M_];

  const int bf = blockIdx.x;
  const int b = bf / F_, f = bf % F_;
  const int tid = threadIdx.x;

  const float* Ub   = U + (long)b * E_ * M_;
  const float* grow = g + (long)bf * E_;

  float acc[M_];
  for (int m = 0; m < M_; ++m) acc[m] = 0.f;
  for (int e = tid; e < E_; e += 256) {
    const float gv = grow[e];
    const float* ur = Ub + (long)e * M_;
    for (int m = 0; m < M_; ++m) acc[m] += gv * ur[m];
  }
  for (int m = 0; m < M_; ++m) red[tid * M_ + m] = acc[m];
  __syncthreads();
  for (int s = 128; s > 0; s >>= 1) {
    if (tid < s)
      for (int m = 0; m < M_; ++m) red[tid * M_ + m] += red[(tid + s) * M_ + m];
    __syncthreads();
  }
  if (tid < M_) yU[tid] = red[tid];
  __syncthreads();

  if (tid < KSTEPS * 32) {
    const int k = tid >> 5, j = tid & 31;
    if (j < M_) {
      const float dtk = DT_ * (float)(k + 1);
      const float* om = omega + (long)bf * M_;
      float val;
      if (j < 2 * NC_) {
        const int c = j >> 1;
        const float theta = om[2 * c] * dtk;
        const float sc = expf(dtk);
        const float co = sc * cosf(theta), si = sc * sinf(theta);
        const float y0 = yU[2 * c], y1 = yU[2 * c + 1];
        val = (j & 1) ? (si * y0 + co * y1) : (co * y0 - si * y1);
      } else {
        val = yU[j] * expf(om[j] * dtk);
      }
      yn[k][j] = val;
    }
  }
  __syncthreads();

  for (int e = tid; e < E_; e += 256) {
    const float* ur = Ub + (long)e * M_;
    float u[M_];
    for (int m = 0; m < M_; ++m) u[m] = ur[m];
    for (int k = 0; k < KSTEPS; ++k) {
      float d = 0.f;
      for (int m = 0; m < M_; ++m) d += u[m] * yn[k][m];
      g_next[(((long)k * B_ + b) * F_ + f) * E_ + e] = d;
    }
  }
}

// ---------------------------------------------------------------------------
extern "C" void kernel_launch(void* const* d_in, const int* in_sizes, int n_in,
                              void* d_out, int out_size, void* d_ws, size_t ws_size,
                              hipStream_t stream)
{
  (void)in_sizes; (void)n_in; (void)out_size; (void)ws_size;
  const float* x      = (const float*)d_in[0];
  const float* em_nfW = (const float*)d_in[1];
  const float* em_nfb = (const float*)d_in[2];
  const float* em_eW1 = (const float*)d_in[3];
  const float* em_eb1 = (const float*)d_in[4];
  const float* em_eW2 = (const float*)d_in[5];
  const float* em_eb2 = (const float*)d_in[6];
  const float* eeW1   = (const float*)d_in[7];
  const float* eeb1   = (const float*)d_in[8];
  const float* eeW2   = (const float*)d_in[9];
  const float* eeb2   = (const float*)d_in[10];
  const float* dW1    = (const float*)d_in[11];
  const float* db1    = (const float*)d_in[12];
  const float* dW2    = (const float*)d_in[13];
  const float* db2    = (const float*)d_in[14];
  const float* epW1   = (const float*)d_in[15];
  const float* epb1   = (const float*)d_in[16];
  const float* epW2   = (const float*)d_in[17];
  const float* epb2   = (const float*)d_in[18];
  const float* evW1   = (const float*)d_in[19];
  const float* evb1   = (const float*)d_in[20];
  const float* evW2   = (const float*)d_in[21];
  const float* evb2   = (const float*)d_in[22];
  const float* evW3   = (const float*)d_in[23];
  const float* evb3   = (const float*)d_in[24];
  const float* evW4   = (const float*)d_in[25];
  const float* evb4   = (const float*)d_in[26];

  float* out = (float*)d_out;
  float* ym = (float*)d_ws;                       // (M,B*N,ELAT)
  float* z1 = ym + (long)M_ * B_ * N_ * ELAT_;    // (256,128)
  float* z2 = z1 + 256 * 128;
  float* z3 = z2 + 256 * 128;

  float* y_o  = out;                                // (B,F,N,LAT)
  float* g_o  = y_o  + (long)B_ * F_ * N_ * LAT_;   // (B,F,E)
  float* gn_o = g_o  + (long)B_ * F_ * E_;          // (K,B,F,E)
  float* U_o  = gn_o + (long)KSTEPS * B_ * F_ * E_; // (B,E,M)
  float* om_o = U_o  + (long)B_ * E_ * M_;          // (B,F,M)
  float* xr_o = om_o + (long)B_ * F_ * M_;          // (B,N,T)

  // 1) ym[m] = x @ em_nfW[m] + em_nfb[m]   (dominant GEMM, 5.4 GF)
  gemm_wmma_wide<4><<<dim3(1, (B_ * N_) / 16, M_), 32, 0, stream>>>(
      x, T_, 0L, em_nfW, ELAT_, (long)T_ * ELAT_, em_nfb, (long)ELAT_,
      ym, ELAT_, (long)B_ * N_ * ELAT_, T_, 0);

  // 2) U[b,e,m] = edgeMLP_m(pearson(ym)[triu])
  pearson_edge_kernel<ELAT_, true><<<M_ * B_, 256, 0, stream>>>(
      ym, em_eW1, em_eb1, em_eW2, em_eb2, U_o);

  // 3) y, x_recon (fused encoder/decoder)
  encdec_kernel<<<B_ * F_, 256, 0, stream>>>(
      x, eeW1, eeb1, eeW2, eeb2, dW1, db1, dW2, db2, y_o, xr_o);

  // 4) g[b,f,e] = edgeMLP(pearson(y)[triu])
  pearson_edge_kernel<LAT_, false><<<B_ * F_, 256, 0, stream>>>(
      y_o, epW1, epb1, epW2, epb2, g_o);

  // 5) z-path: (256,4096)@(4096,128) -> 2x(128,128) -> (128,20)
  gemm_wmma_wide<4><<<dim3(2, 16, 1), 32, 0, stream>>>(
      y_o, N_ * LAT_, 0L, evW1, 128, 0L, evb1, 0L, z1, 128, 0L,
      N_ * LAT_, 1);
  gemm_wmma_wide<4><<<dim3(2, 16, 1), 32, 0, stream>>>(
      z1, 128, 0L, evW2, 128, 0L, evb2, 0L, z2, 128, 0L, 128, 1);
  gemm_wmma_wide<4><<<dim3(2, 16, 1), 32, 0, stream>>>(
      z2, 128, 0L, evW3, 128, 0L, evb3, 0L, z3, 128, 0L, 128, 1);
  gemm_wmma<<<dim3(2, 16, 1), 32, 0, stream>>>(
      z3, 128, 0L, evW4, M_, 0L, evb4, 0L, om_o, M_, 0L, M_, 128, 0);

  // 6) g_next for k = 1..4
  vm_kernel<<<B_ * F_, 256, 0, stream>>>(U_o, g_o, om_o, gn_o);
}